// ConceptEmbedModel_65695819759690
// MI455X (gfx1250) — compile-verified
//
#include <hip/hip_runtime.h>
#include <math.h>

typedef __attribute__((ext_vector_type(2))) float v2f;
typedef __attribute__((ext_vector_type(4))) float v4f;
typedef __attribute__((ext_vector_type(8))) float v8f;

#define EMBED_DIM    200
#define VOCAB        100000
#define NUM_CONCEPTS 50000
#define TOTAL_ANC    800000
#define BATCH        4096

#define NTILES  (NUM_CONCEPTS / 16)   // 3125 N tiles
#define MTILES  (BATCH / 16)          // 256 M tiles (power of 2)
#define NSTRIPS ((NTILES + 3) / 4)    // 782 strips of 4 N-tiles per wave

// ---------------------------------------------------------------------------
// Kernel 0: zero the H accumulator in workspace (f32x4 stores).
// ---------------------------------------------------------------------------
__global__ void k_zero(float* __restrict__ p, int n4) {
    int i = blockIdx.x * blockDim.x + threadIdx.x;
    if (i < n4) {
        v4f z = {};
        ((v4f*)p)[i] = z;
    }
}

// ---------------------------------------------------------------------------
// Kernel 1: ragged gather + segment-sum into H[50000, 200].
// segment_ids is sorted, so each block walks 32 consecutive flat entries,
// accumulates same-segment runs in registers, and issues one f32 global
// atomic per run boundary per dim. Thread d (< 200) owns one embed dim.
// ---------------------------------------------------------------------------
__global__ void k_segsum(const float* __restrict__ W,
                         const int* __restrict__ flat_idx,
                         const int* __restrict__ seg,
                         float* __restrict__ H) {
    const int d = threadIdx.x;
    if (d >= EMBED_DIM) return;
    const int t0 = blockIdx.x * 32;

    int   cur = -1;
    float acc = 0.0f;
    #pragma unroll 1
    for (int i = 0; i < 32; ++i) {
        int t = t0 + i;
        if (t >= TOTAL_ANC) break;
        int   s = seg[t];            // uniform across the block
        int   e = flat_idx[t];       // uniform across the block
        float w = W[(size_t)e * EMBED_DIM + d];
        if (s != cur) {
            if (cur >= 0) unsafeAtomicAdd(&H[(size_t)cur * EMBED_DIM + d], acc);
            cur = s;
            acc = 0.0f;
        }
        acc += w;
    }
    if (cur >= 0) unsafeAtomicAdd(&H[(size_t)cur * EMBED_DIM + d], acc);
}

// ---------------------------------------------------------------------------
// Kernel 2: logits = x[4096,200] * H^T  via V_WMMA_F32_16X16X4_F32.
// One wave computes a 16(M) x 64(N) strip = 4 accumulator tiles; the A
// fragment is reused across the 4 WMMAs per K-step. K=200 -> 50 steps.
// Fragment layout (32-bit, 16x4 A / 4x16 B):
//   lanes 0-15 : M/N = lane,     VGPR0=K0, VGPR1=K1
//   lanes 16-31: M/N = lane-16,  VGPR0=K2, VGPR1=K3
// -> each lane loads one float2 at column (k0 + 2*(lane>>4)).
//
// Uniformity: wave id is forced into an SGPR via readfirstlane so all tile
// arithmetic and guards are scalar -> EXEC stays all-ones around WMMA.
// Tail strip (ns==781 has only 1 valid N-tile): B row indices are CLAMPED so
// loads stay in bounds; invalid tiles compute garbage but are never stored.
// K-loop is branch-free and software-pipelined (prefetch next fragments
// before issuing current WMMAs).
// ---------------------------------------------------------------------------
__global__ void k_gemm(const float* __restrict__ X,
                       const float* __restrict__ H,
                       float* __restrict__ OUT) {
    const int lane = threadIdx.x & 31;
    const int wid  = __builtin_amdgcn_readfirstlane(blockIdx.x * 8 + (threadIdx.x >> 5));
    const int m    = wid & (MTILES - 1);   // M tile index (scalar)
    const int ns   = wid >> 8;             // N strip index (scalar)
    if (ns >= NSTRIPS) return;             // scalar branch, whole wave

    const int n0 = ns * 64;
    const int nv = (NTILES - ns * 4) >= 4 ? 4 : (NTILES - ns * 4); // scalar

    const int kh   = (lane >> 4) << 1;           // 0 or 2
    const int rowA = m * 16 + (lane & 15);
    const float* pA = X + (size_t)rowA * EMBED_DIM + kh;

    const float* pB[4];
    #pragma unroll
    for (int j = 0; j < 4; ++j) {
        int rowB = n0 + j * 16 + (lane & 15);
        rowB = rowB < NUM_CONCEPTS ? rowB : (NUM_CONCEPTS - 1); // clamp tail
        pB[j] = H + (size_t)rowB * EMBED_DIM + kh;
    }

    v8f acc[4] = {};

    // software pipeline: preload K-step 0
    v2f a = *(const v2f*)pA;
    v2f b[4];
    #pragma unroll
    for (int j = 0; j < 4; ++j) b[j] = *(const v2f*)pB[j];

    #pragma unroll 2
    for (int k0 = 4; k0 < EMBED_DIM; k0 += 4) {
        v2f an = *(const v2f*)(pA + k0);
        v2f bn[4];
        #pragma unroll
        for (int j = 0; j < 4; ++j) bn[j] = *(const v2f*)(pB[j] + k0);

        #pragma unroll
        for (int j = 0; j < 4; ++j) {
            acc[j] = __builtin_amdgcn_wmma_f32_16x16x4_f32(
                false, a, false, b[j], (short)0, acc[j], false, false);
        }
        a = an;
        #pragma unroll
        for (int j = 0; j < 4; ++j) b[j] = bn[j];
    }
    #pragma unroll
    for (int j = 0; j < 4; ++j) {
        acc[j] = __builtin_amdgcn_wmma_f32_16x16x4_f32(
            false, a, false, b[j], (short)0, acc[j], false, false);
    }

    // C/D layout: VGPR r -> M = r (lanes 0-15) or 8+r (lanes 16-31), N = lane%16
    const int rowBase = m * 16 + ((lane >> 4) << 3);
    const int col     = lane & 15;
    #pragma unroll
    for (int j = 0; j < 4; ++j) {
        if (j < nv) {                       // scalar guard
            int c = n0 + j * 16 + col;
            #pragma unroll
            for (int r = 0; r < 8; ++r) {
                OUT[(size_t)(rowBase + r) * NUM_CONCEPTS + c] = acc[j][r];
            }
        }
    }
}

// ---------------------------------------------------------------------------
// Kernel 3: per-row online softmax statistics (max + sum of exp), one block
// per batch row, LDS tree merge of (m, s) pairs. Stores max and 1/sum.
// ---------------------------------------------------------------------------
__global__ void k_rowstats(const float* __restrict__ L,
                           float* __restrict__ rowm,
                           float* __restrict__ rinv) {
    __shared__ float sm[256];
    __shared__ float ss[256];
    const int r   = blockIdx.x;
    const int tid = threadIdx.x;
    const float* row = L + (size_t)r * NUM_CONCEPTS;

    float m = -INFINITY, s = 0.0f;
    for (int c = tid; c < NUM_CONCEPTS; c += 256) {
        float v = row[c];
        if (v > m) {
            s = s * __expf(m - v) + 1.0f;
            m = v;
        } else {
            s += __expf(v - m);
        }
    }
    sm[tid] = m; ss[tid] = s;
    __syncthreads();
    for (int off = 128; off > 0; off >>= 1) {
        if (tid < off) {
            float m1 = sm[tid],       s1 = ss[tid];
            float m2 = sm[tid + off], s2 = ss[tid + off];
            float M  = fmaxf(m1, m2);
            ss[tid] = s1 * __expf(m1 - M) + s2 * __expf(m2 - M);
            sm[tid] = M;
        }
        __syncthreads();
    }
    if (tid == 0) {
        rowm[r] = sm[0];
        rinv[r] = 1.0f / ss[0];
    }
}

// ---------------------------------------------------------------------------
// Kernel 4: in-place exp-normalize, float4 vectorized (50000 % 4 == 0 so
// rows stay aligned to float4 boundaries).
// ---------------------------------------------------------------------------
__global__ void k_softmax(float* __restrict__ OUT,
                          const float* __restrict__ rowm,
                          const float* __restrict__ rinv) {
    const int n4 = BATCH * (NUM_CONCEPTS / 4);
    int i = blockIdx.x * blockDim.x + threadIdx.x;
    if (i >= n4) return;
    int b = i / (NUM_CONCEPTS / 4);
    float m   = rowm[b];
    float inv = rinv[b];
    v4f v = ((v4f*)OUT)[i];
    v.x = __expf(v.x - m) * inv;
    v.y = __expf(v.y - m) * inv;
    v.z = __expf(v.z - m) * inv;
    v.w = __expf(v.w - m) * inv;
    ((v4f*)OUT)[i] = v;
}

// ---------------------------------------------------------------------------
extern "C" void kernel_launch(void* const* d_in, const int* in_sizes, int n_in,
                              void* d_out, int out_size, void* d_ws, size_t ws_size,
                              hipStream_t stream) {
    const float* x        = (const float*)d_in[0]; // [4096,1,200]
    const float* embed_w  = (const float*)d_in[1]; // [100000,200]
    const int*   flat_idx = (const int*)d_in[2];   // [800000]
    const int*   seg_ids  = (const int*)d_in[3];   // [800000] sorted
    (void)in_sizes; (void)n_in; (void)out_size; (void)ws_size;

    // workspace layout: H (40MB) | rowm (16KB) | rinv (16KB)
    float* H    = (float*)d_ws;
    float* rowm = (float*)((char*)d_ws + (size_t)NUM_CONCEPTS * EMBED_DIM * sizeof(float));
    float* rinv = rowm + BATCH;
    float* out  = (float*)d_out;

    // 0) zero H
    {
        int n4 = (NUM_CONCEPTS * EMBED_DIM) / 4; // 2.5M float4
        k_zero<<<(n4 + 255) / 256, 256, 0, stream>>>(H, n4);
    }
    // 1) segment sum (gather)
    k_segsum<<<TOTAL_ANC / 32, 256, 0, stream>>>(embed_w, flat_idx, seg_ids, H);

    // 2) WMMA f32 GEMM: logits -> d_out
    {
        int totalWaves = MTILES * NSTRIPS;      // 256 * 782 = 200192
        int blocks     = totalWaves / 8;        // 25024 (8 waves / block)
        k_gemm<<<blocks, 256, 0, stream>>>(x, H, out);
    }
    // 3) row stats
    k_rowstats<<<BATCH, 256, 0, stream>>>(out, rowm, rinv);

    // 4) normalize in place
    {
        int n4 = BATCH * (NUM_CONCEPTS / 4);    // 51.2M float4
        k_softmax<<<(n4 + 255) / 256, 256, 0, stream>>>(out, rowm, rinv);
    }
}